// RotatedIoU3DLoss_66769561583663
// MI455X (gfx1250) — compile-verified
//
#include <hip/hip_runtime.h>
#include <math.h>

#define EPS_IN 1e-6f
#define EPS_IOU 1e-8f

typedef __attribute__((ext_vector_type(2))) float v2f;
typedef __attribute__((ext_vector_type(8))) float v8f;

__device__ __forceinline__ bool point_in_box(float x, float y,
                                             const float* cx, const float* cy) {
  // a = corner0, b = corner1, d = corner3 (matches reference)
  float ax = cx[0], ay = cy[0];
  float abx = cx[1] - ax, aby = cy[1] - ay;
  float adx = cx[3] - ax, ady = cy[3] - ay;
  float apx = x - ax,     apy = y - ay;
  float pab = (apx * abx + apy * aby) / (abx * abx + aby * aby);
  float pad = (apx * adx + apy * ady) / (adx * adx + ady * ady);
  return (pab > -EPS_IN) && (pab < 1.0f + EPS_IN) &&
         (pad > -EPS_IN) && (pad < 1.0f + EPS_IN);
}

__global__ void __launch_bounds__(256)
iou3d_loss_kernel(const float* __restrict__ pred,
                  const float* __restrict__ target,
                  float* __restrict__ partial, int n) {
  int i = blockIdx.x * blockDim.x + threadIdx.x;
  float loss = 0.0f;
  if (i < n) {
    const float* P = pred   + (size_t)i * 7;
    const float* T = target + (size_t)i * 7;
    float px = P[0], py = P[1], pz = P[2], pw = P[3], pl = P[4], ph = P[5], pa = P[6];
    float qx = T[0], qy = T[1], qz = T[2], qw = T[3], ql = T[4], qh = T[5], qa = T[6];

    const float TXc[4] = {0.5f, -0.5f, -0.5f, 0.5f};
    const float TYc[4] = {0.5f, 0.5f, -0.5f, -0.5f};
    float c1x[4], c1y[4], c2x[4], c2y[4];
    float cp = cosf(pa), sp = sinf(pa);
    float cq = cosf(qa), sq = sinf(qa);
#pragma unroll
    for (int k = 0; k < 4; ++k) {
      float x4 = TXc[k] * pw, y4 = TYc[k] * pl;
      c1x[k] = x4 * cp - y4 * sp + px;
      c1y[k] = x4 * sp + y4 * cp + py;
      float x42 = TXc[k] * qw, y42 = TYc[k] * ql;
      c2x[k] = x42 * cq - y42 * sq + qx;
      c2y[k] = x42 * sq + y42 * cq + qy;
    }

    // 24 candidate vertices: 4 pred corners, 4 target corners, 16 edge intersections
    float vx[24], vy[24];
    bool  vm[24];
#pragma unroll
    for (int k = 0; k < 4; ++k) {
      vx[k] = c1x[k]; vy[k] = c1y[k];
      vm[k] = point_in_box(c1x[k], c1y[k], c2x, c2y);
      vx[4 + k] = c2x[k]; vy[4 + k] = c2y[k];
      vm[4 + k] = point_in_box(c2x[k], c2y[k], c1x, c1y);
    }
    int idx = 8;
#pragma unroll
    for (int a = 0; a < 4; ++a) {
      float p1x = c1x[a], p1y = c1y[a];
      float rx = c1x[(a + 1) & 3] - p1x, ry = c1y[(a + 1) & 3] - p1y;
#pragma unroll
      for (int b = 0; b < 4; ++b) {
        float q1x = c2x[b], q1y = c2y[b];
        float sx = c2x[(b + 1) & 3] - q1x, sy = c2y[(b + 1) & 3] - q1y;
        float den  = rx * sy - ry * sx;
        float dens = (den == 0.0f) ? 1.0f : den;
        float wx = q1x - p1x, wy = q1y - p1y;
        float t = (wx * sy - wy * sx) / dens;
        float u = (wx * ry - wy * rx) / dens;
        bool m = (den != 0.0f) && (t > 0.0f) && (t < 1.0f) && (u > 0.0f) && (u < 1.0f);
        vx[idx] = m ? (p1x + t * rx) : 0.0f;
        vy[idx] = m ? (p1y + t * ry) : 0.0f;
        vm[idx] = m;
        ++idx;
      }
    }

    // polygon area via angle sort around masked centroid (replicates reference)
    float nv = 0.0f, sx = 0.0f, sy = 0.0f;
#pragma unroll
    for (int k = 0; k < 24; ++k) {
      float mk = vm[k] ? 1.0f : 0.0f;
      nv += mk; sx += vx[k] * mk; sy += vy[k] * mk;
    }
    float nvv = fmaxf(nv, 1.0f);
    float mx = sx / nvv, my = sy / nvv;
    float ang[24];
#pragma unroll
    for (int k = 0; k < 24; ++k) {
      vx[k] -= mx; vy[k] -= my;
      ang[k] = vm[k] ? atan2f(vy[k], vx[k]) : 1.0e9f;
    }
    // stable insertion sort (matches jnp.argsort stability)
    for (int k = 1; k < 24; ++k) {
      float a = ang[k], xx = vx[k], yy = vy[k];
      bool m = vm[k];
      int j = k - 1;
      while (j >= 0 && ang[j] > a) {
        ang[j + 1] = ang[j]; vx[j + 1] = vx[j]; vy[j + 1] = vy[j]; vm[j + 1] = vm[j];
        --j;
      }
      ang[j + 1] = a; vx[j + 1] = xx; vy[j + 1] = yy; vm[j + 1] = m;
    }
    float v0x = vx[0], v0y = vy[0];
#pragma unroll
    for (int k = 0; k < 24; ++k) {
      if (!vm[k]) { vx[k] = v0x; vy[k] = v0y; }
    }
    float area2 = 0.0f;
#pragma unroll
    for (int k = 0; k < 24; ++k) {
      int kn = (k + 1) % 24;
      area2 += vx[k] * vy[kn] - vy[k] * vx[kn];
    }
    float inter_area = 0.5f * fabsf(area2);

    float zt = fminf(pz + 0.5f * ph, qz + 0.5f * qh);
    float zb = fmaxf(pz - 0.5f * ph, qz - 0.5f * qh);
    float inter_vol = inter_area * fmaxf(zt - zb, 0.0f);
    float v1 = pw * pl * ph, v2 = qw * ql * qh;
    float iou = inter_vol / (v1 + v2 - inter_vol + EPS_IOU);
    loss = 1.0f - iou;
  }

  __shared__ float red[256];
  red[threadIdx.x] = loss;
  __syncthreads();
#pragma unroll
  for (int s = 128; s > 0; s >>= 1) {
    if (threadIdx.x < s) red[threadIdx.x] += red[threadIdx.x + s];
    __syncthreads();
  }
  if (threadIdx.x == 0) partial[blockIdx.x] = red[0];
}

// Final reduction: per-lane strided sums -> LDS -> wave0 32 lanes ->
// V_WMMA_F32_16X16X4_F32 with all-ones B: D[m][n] = rowsum_A(m), so
// summing one D column over rows (8 regs at lane n plus 8 at lane n+16)
// equals the exact sum of all 64 A elements regardless of lane mapping.
__global__ void __launch_bounds__(256)
final_reduce_kernel(const float* __restrict__ partial, int nPartial,
                    float* __restrict__ out, float invN) {
  __shared__ float red[256];
  float acc = 0.0f;
  for (int i = threadIdx.x; i < nPartial; i += 256) acc += partial[i];
  red[threadIdx.x] = acc;
  __syncthreads();
  if (threadIdx.x < 32) {               // wave-uniform branch: EXEC all ones in wave 0
    float l = 0.0f;
#pragma unroll
    for (int k = 0; k < 8; ++k) l += red[threadIdx.x + 32 * k];
    v2f A; A.x = l;    A.y = 0.0f;      // 32 lane values occupy unique (M,K) slots
    v2f B; B.x = 1.0f; B.y = 1.0f;      // all-ones: layout-independent row sums
    v8f C = {};
    C = __builtin_amdgcn_wmma_f32_16x16x4_f32(
        /*neg_a=*/false, A, /*neg_b=*/false, B,
        /*c_mod=*/(short)0, C, /*reuse_a=*/false, /*reuse_b=*/false);
    float s8 = C[0] + C[1] + C[2] + C[3] + C[4] + C[5] + C[6] + C[7];
    float tot = s8 + __shfl_xor(s8, 16, 32);   // rows 0-7 (lane n) + rows 8-15 (lane n+16)
    if (threadIdx.x == 0) out[0] = tot * invN;
  }
}

extern "C" void kernel_launch(void* const* d_in, const int* in_sizes, int n_in,
                              void* d_out, int out_size, void* d_ws, size_t ws_size,
                              hipStream_t stream) {
  const float* pred   = (const float*)d_in[0];
  const float* target = (const float*)d_in[1];
  int n = in_sizes[0] / 7;
  int blocks = (n + 255) / 256;
  float* partial = (float*)d_ws;
  iou3d_loss_kernel<<<blocks, 256, 0, stream>>>(pred, target, partial, n);
  final_reduce_kernel<<<1, 256, 0, stream>>>(partial, blocks, (float*)d_out,
                                             1.0f / (float)n);
}